// HRGCN_35923106463911
// MI455X (gfx1250) — compile-verified
//
#include <hip/hip_runtime.h>
#include <math.h>

#define NN 100000
#define EE 1000000
#define BB 16
#define TT 7
#define RR 4
#define DD 128
#define HH 256
#define OO 128

typedef __attribute__((ext_vector_type(2))) float v2f;
typedef __attribute__((ext_vector_type(8))) float v8f;
typedef int v4i __attribute__((vector_size(16)));

#if defined(__has_builtin)
#if __has_builtin(__builtin_amdgcn_global_load_async_to_lds_b128) && \
    __has_builtin(__builtin_amdgcn_s_wait_asynccnt)
#define HAVE_ASYNC_LDS 1
#endif
#endif

__device__ __forceinline__ v8f wmma4(v2f a, v2f b, v8f c) {
  // D = A(16x4) * B(4x16) + C, fp32 exact
  return __builtin_amdgcn_wmma_f32_16x16x4_f32(
      /*neg_a=*/false, a, /*neg_b=*/false, b,
      /*c_mod=*/(short)0, c, /*reuse_a=*/false, /*reuse_b=*/false);
}

// ---------------------------------------------------------------------------
// Encoder: h0[n,:] = x[n,:] @ W_nt[node_type[n]]   (one block per node)
// x row staged in LDS via async global->LDS copy (gfx1250 path).
// ---------------------------------------------------------------------------
__global__ void encoder_kernel(const float* __restrict__ x,
                               const float* __restrict__ Wnt,
                               const int* __restrict__ ntype,
                               float* __restrict__ h0) {
  __shared__ __align__(16) float xs[DD];
  const int n = blockIdx.x;
  const int j = threadIdx.x;  // 0..127
#ifdef HAVE_ASYNC_LDS
  if (j < DD / 4) {  // 32 lanes x 16B = 512B row
    typedef __attribute__((address_space(1))) v4i gv4i;
    typedef __attribute__((address_space(3))) v4i sv4i;
    gv4i* g = (gv4i*)(void*)(x + (size_t)n * DD + j * 4);
    sv4i* l = (sv4i*)(void*)(xs + j * 4);
    __builtin_amdgcn_global_load_async_to_lds_b128(g, l, 0, 0);
  }
  __builtin_amdgcn_s_wait_asynccnt(0);
  __syncthreads();
#else
  xs[j] = x[(size_t)n * DD + j];
  __syncthreads();
#endif
  const int t = ntype[n];
  const float* __restrict__ W = Wnt + (size_t)t * DD * DD;
  float acc = 0.0f;
#pragma unroll 8
  for (int k = 0; k < DD; ++k) acc = fmaf(xs[k], W[k * DD + j], acc);
  h0[(size_t)n * DD + j] = acc;
}

// ---------------------------------------------------------------------------
// Per-(dst,rel) in-degree counts. Graph is fixed: computed once, reused 3x.
// ---------------------------------------------------------------------------
__global__ void count_kernel(const int* __restrict__ edst,
                             const int* __restrict__ etyp,
                             float* __restrict__ cnt) {
  const int stride = gridDim.x * blockDim.x;
  for (int e = blockIdx.x * blockDim.x + threadIdx.x; e < EE; e += stride)
    atomicAdd(cnt + edst[e] * RR + etyp[e], 1.0f);
}

// ---------------------------------------------------------------------------
// Layer-1 aggregate-then-transform scatter with mean folded in:
//   agg[(dst*R+rel)*Dh + j] += h[src*Dh + j] / max(cnt[dst*R+rel],1)
// ---------------------------------------------------------------------------
__global__ void scatter_in_scaled(const float* __restrict__ h,
                                  const float* __restrict__ cnt,
                                  const int* __restrict__ esrc,
                                  const int* __restrict__ edst,
                                  const int* __restrict__ etyp,
                                  float* __restrict__ agg, int shift) {
  const long total = (long)EE << shift;
  const int mask = (1 << shift) - 1;
  const long stride = (long)gridDim.x * blockDim.x;
  for (long idx = (long)blockIdx.x * blockDim.x + threadIdx.x; idx < total;
       idx += stride) {
    const int e = (int)(idx >> shift);
    const int j = (int)idx & mask;
    const int seg = edst[e] * RR + etyp[e];
    const float w = 1.0f / fmaxf(cnt[seg], 1.0f);
    atomicAdd(agg + (((long)seg) << shift) + j,
              w * h[(((long)esrc[e]) << shift) + j]);
  }
}

// ---------------------------------------------------------------------------
// Transform-first scatter (layers 2/3): relations merge on the output side,
// destination is N*Dho (<=100MB -> L2-resident atomics):
//   out[dst*Dho + j] += Y[(src*R+rel)*Dho + j] / max(cnt[dst*R+rel],1)
// ---------------------------------------------------------------------------
__global__ void scatter_out_scaled(const float* __restrict__ Y,
                                   const float* __restrict__ cnt,
                                   const int* __restrict__ esrc,
                                   const int* __restrict__ edst,
                                   const int* __restrict__ etyp,
                                   float* __restrict__ out, int shift) {
  const long total = (long)EE << shift;
  const int mask = (1 << shift) - 1;
  const long stride = (long)gridDim.x * blockDim.x;
  for (long idx = (long)blockIdx.x * blockDim.x + threadIdx.x; idx < total;
       idx += stride) {
    const int e = (int)(idx >> shift);
    const int j = (int)idx & mask;
    const int r = etyp[e];
    const int d = edst[e];
    const float w = 1.0f / fmaxf(cnt[d * RR + r], 1.0f);
    atomicAdd(out + (((long)d) << shift) + j,
              w * Y[(((long)esrc[e] * RR + r) << shift) + j]);
  }
}

__global__ void relu_kernel(float* __restrict__ p, long n) {
  const long stride = (long)gridDim.x * blockDim.x;
  for (long i = (long)blockIdx.x * blockDim.x + threadIdx.x; i < n; i += stride)
    p[i] = fmaxf(p[i], 0.0f);
}

// ---------------------------------------------------------------------------
// fp32 WMMA GEMM: C[M,N] = (acc? C : bias : 0) + A[M,K] @ B[K,N]; opt. ReLU.
// One wave per 32x64 output tile (2 M-subtiles x 4 N-subtiles => 8 WMMAs per
// k-step; B fragments reused across both M-subtiles).
// C has row stride ldc (>= Ncols) so strided Y[(n*R+r)*Dho] can be written.
// A frag (16x4 f32): lanes 0-15 hold K={0,1}, lanes 16-31 hold K={2,3}.
// C/D (16x16 f32): VGPR v -> M=v (lanes 0-15), M=v+8 (lanes 16-31).
// ---------------------------------------------------------------------------
__global__ void gemm_wmma(const float* __restrict__ A,
                          const float* __restrict__ B,
                          const float* __restrict__ bias,
                          float* __restrict__ C,
                          int K, int Ncols, int ldc, int accumulate, int relu) {
  const int lane = threadIdx.x & 31;
  const int m0 = blockIdx.y * 32;
  const int n0 = blockIdx.x * 64;
  const int mr = lane & 15;           // M-row (A) / N-col (B,C) sub-index
  const int ks = (lane >> 4) << 1;    // K sub-offset: 0 or 2
  const int crow = (lane >> 4) << 3;  // C row offset: 0 or 8

  v8f acc[2][4];
  if (accumulate) {
#pragma unroll
    for (int mt = 0; mt < 2; ++mt)
#pragma unroll
      for (int v = 0; v < 8; ++v) {
        const size_t base = (size_t)(m0 + mt * 16 + v + crow) * ldc + n0 + mr;
        acc[mt][0][v] = C[base];
        acc[mt][1][v] = C[base + 16];
        acc[mt][2][v] = C[base + 32];
        acc[mt][3][v] = C[base + 48];
      }
  } else if (bias) {
    float bv[4];
#pragma unroll
    for (int t = 0; t < 4; ++t) bv[t] = bias[n0 + t * 16 + mr];
#pragma unroll
    for (int mt = 0; mt < 2; ++mt)
#pragma unroll
      for (int t = 0; t < 4; ++t)
#pragma unroll
        for (int v = 0; v < 8; ++v) acc[mt][t][v] = bv[t];
  } else {
#pragma unroll
    for (int mt = 0; mt < 2; ++mt)
#pragma unroll
      for (int t = 0; t < 4; ++t) acc[mt][t] = (v8f){};
  }

  const float* __restrict__ Ar0 = A + (size_t)(m0 + mr) * K;
  const float* __restrict__ Ar1 = Ar0 + (size_t)16 * K;
  for (int k0 = 0; k0 < K; k0 += 4) {
    if (k0 + 64 < K) {  // uniform branch: EXEC unchanged (WMMA-safe)
      __builtin_prefetch(Ar0 + k0 + 64, 0, 3);
      __builtin_prefetch(Ar1 + k0 + 64, 0, 3);
    }
    v2f a0, a1;
    a0.x = Ar0[k0 + ks];
    a0.y = Ar0[k0 + ks + 1];
    a1.x = Ar1[k0 + ks];
    a1.y = Ar1[k0 + ks + 1];
    const float* __restrict__ Bp = B + (size_t)(k0 + ks) * Ncols + n0 + mr;
    const float* __restrict__ Bq = Bp + Ncols;
    v2f b;
    b.x = Bp[0];
    b.y = Bq[0];
    acc[0][0] = wmma4(a0, b, acc[0][0]);
    acc[1][0] = wmma4(a1, b, acc[1][0]);
    b.x = Bp[16];
    b.y = Bq[16];
    acc[0][1] = wmma4(a0, b, acc[0][1]);
    acc[1][1] = wmma4(a1, b, acc[1][1]);
    b.x = Bp[32];
    b.y = Bq[32];
    acc[0][2] = wmma4(a0, b, acc[0][2]);
    acc[1][2] = wmma4(a1, b, acc[1][2]);
    b.x = Bp[48];
    b.y = Bq[48];
    acc[0][3] = wmma4(a0, b, acc[0][3]);
    acc[1][3] = wmma4(a1, b, acc[1][3]);
  }

#pragma unroll
  for (int mt = 0; mt < 2; ++mt)
#pragma unroll
    for (int v = 0; v < 8; ++v) {
      const size_t base = (size_t)(m0 + mt * 16 + v + crow) * ldc + n0 + mr;
      float r0 = acc[mt][0][v], r1 = acc[mt][1][v];
      float r2 = acc[mt][2][v], r3 = acc[mt][3][v];
      if (relu) {
        r0 = fmaxf(r0, 0.0f);
        r1 = fmaxf(r1, 0.0f);
        r2 = fmaxf(r2, 0.0f);
        r3 = fmaxf(r3, 0.0f);
      }
      C[base] = r0;
      C[base + 16] = r1;
      C[base + 32] = r2;
      C[base + 48] = r3;
    }
}

// ---------------------------------------------------------------------------
// Graph readout
// ---------------------------------------------------------------------------
__global__ void pool_scatter_kernel(const float* __restrict__ h,
                                    const int* __restrict__ gid,
                                    float* __restrict__ pool) {
  const int total = NN * OO;
  const int stride = gridDim.x * blockDim.x;
  for (int idx = blockIdx.x * blockDim.x + threadIdx.x; idx < total;
       idx += stride) {
    const int n = idx >> 7;  // OO == 128
    const int j = idx & 127;
    atomicAdd(pool + gid[n] * OO + j, h[idx]);
  }
}

__global__ void gcount_kernel(const int* __restrict__ gid,
                              float* __restrict__ gcnt) {
  const int stride = gridDim.x * blockDim.x;
  for (int n = blockIdx.x * blockDim.x + threadIdx.x; n < NN; n += stride)
    atomicAdd(gcnt + gid[n], 1.0f);
}

// out layout: [0,16) scores, [16,32) labels(=0), [32,48) methods(=0),
//             [48, 48+B*O) emb
__global__ void finalize_kernel(const float* __restrict__ pool,
                                const float* __restrict__ gcnt,
                                const float* __restrict__ wfc,
                                const float* __restrict__ bfc,
                                float* __restrict__ out) {
  __shared__ float red[OO];
  const int b = blockIdx.x;   // 0..B-1
  const int j = threadIdx.x;  // 0..O-1
  const float c = fmaxf(gcnt[b], 1.0f);
  float e = pool[b * OO + j] / c;
  e = (e > 0.0f) ? e : 0.01f * e;  // leaky_relu(0.01)
  out[48 + b * OO + j] = e;        // emb
  red[j] = e * wfc[j];
  __syncthreads();
  for (int s = OO / 2; s > 0; s >>= 1) {
    if (j < s) red[j] += red[j + s];
    __syncthreads();
  }
  if (j == 0) {
    const float z = red[0] + bfc[0];
    out[b] = 1.0f / (1.0f + expf(-z));  // sigmoid score
    out[16 + b] = 0.0f;                 // labels (int 0 bit pattern)
    out[32 + b] = 0.0f;                 // methods
  }
}

// ---------------------------------------------------------------------------
extern "C" void kernel_launch(void* const* d_in, const int* in_sizes, int n_in,
                              void* d_out, int out_size, void* d_ws,
                              size_t ws_size, hipStream_t stream) {
  const float* x   = (const float*)d_in[0];
  const float* Wnt = (const float*)d_in[1];
  const float* Wr1 = (const float*)d_in[2];
  const float* Ws1 = (const float*)d_in[3];
  const float* b1  = (const float*)d_in[4];
  const float* Wr2 = (const float*)d_in[5];
  const float* Ws2 = (const float*)d_in[6];
  const float* b2  = (const float*)d_in[7];
  const float* Wr3 = (const float*)d_in[8];
  const float* Ws3 = (const float*)d_in[9];
  const float* b3  = (const float*)d_in[10];
  const float* wfc = (const float*)d_in[11];
  const float* bfc = (const float*)d_in[12];
  const int* esrc  = (const int*)d_in[13];
  const int* edst  = (const int*)d_in[14];
  const int* etyp  = (const int*)d_in[15];
  const int* ntyp  = (const int*)d_in[16];
  const int* gid   = (const int*)d_in[17];
  float* out = (float*)d_out;

  // workspace carve-up (floats)
  float* hA   = (float*)d_ws;                // N*H
  float* hB   = hA + (size_t)NN * HH;        // N*H
  float* Y    = hB + (size_t)NN * HH;        // N*R*H (also layer-1 agg)
  float* cnt  = Y + (size_t)NN * RR * HH;    // N*R
  float* pool = cnt + (size_t)NN * RR;       // B*O
  float* gcnt = pool + (size_t)BB * OO;      // B

  // ---- in-degree counts (graph fixed across layers: once) -----------------
  (void)hipMemsetAsync(cnt, 0, (size_t)NN * RR * sizeof(float), stream);
  count_kernel<<<1024, 256, 0, stream>>>(edst, etyp, cnt);

  // ---- encoder: hA = per-type x @ W_nt -------------------------------------
  encoder_kernel<<<NN, DD, 0, stream>>>(x, Wnt, ntyp, hA);

  // ---- layer 1: D -> H, aggregate-then-transform (gather width 128) -------
  (void)hipMemsetAsync(Y, 0, (size_t)NN * RR * DD * sizeof(float), stream);
  scatter_in_scaled<<<4096, 256, 0, stream>>>(hA, cnt, esrc, edst, etyp, Y, 7);
  {
    dim3 grid(HH / 64, NN / 32);
    gemm_wmma<<<grid, 32, 0, stream>>>(hA, Ws1, b1, hB, DD, HH, HH, 0, 0);
    gemm_wmma<<<grid, 32, 0, stream>>>(Y, Wr1, nullptr, hB, RR * DD, HH, HH, 1, 1);
  }

  // ---- layer 2: H -> H, transform-first (merge relations on output) -------
  {
    dim3 grid(HH / 64, NN / 32);
    for (int r = 0; r < RR; ++r)
      gemm_wmma<<<grid, 32, 0, stream>>>(hB, Wr2 + (size_t)r * HH * HH, nullptr,
                                         Y + (size_t)r * HH, HH, HH, RR * HH, 0, 0);
    gemm_wmma<<<grid, 32, 0, stream>>>(hB, Ws2, b2, hA, HH, HH, HH, 0, 0);
  }
  scatter_out_scaled<<<4096, 256, 0, stream>>>(Y, cnt, esrc, edst, etyp, hA, 8);
  relu_kernel<<<2048, 256, 0, stream>>>(hA, (long)NN * HH);

  // ---- layer 3: H -> O, transform-first (gather/scatter width 128) --------
  {
    dim3 grid(OO / 64, NN / 32);
    for (int r = 0; r < RR; ++r)
      gemm_wmma<<<grid, 32, 0, stream>>>(hA, Wr3 + (size_t)r * HH * OO, nullptr,
                                         Y + (size_t)r * OO, HH, OO, RR * OO, 0, 0);
    gemm_wmma<<<grid, 32, 0, stream>>>(hA, Ws3, b3, hB, HH, OO, OO, 0, 0);
  }
  scatter_out_scaled<<<4096, 256, 0, stream>>>(Y, cnt, esrc, edst, etyp, hB, 7);

  // ---- readout -------------------------------------------------------------
  (void)hipMemsetAsync(pool, 0, (size_t)(BB * OO + BB) * sizeof(float), stream);
  pool_scatter_kernel<<<4096, 256, 0, stream>>>(hB, gid, pool);
  gcount_kernel<<<512, 256, 0, stream>>>(gid, gcnt);
  finalize_kernel<<<BB, OO, 0, stream>>>(pool, gcnt, wfc, bfc, out);
}